// BiLSTM_CRF_71141838291390
// MI455X (gfx1250) — compile-verified
//
#include <hip/hip_runtime.h>
#include <hip/hip_bf16.h>

// ---------------- problem constants (match reference) ----------------
#define S_LEN 4096
#define E_DIM 256
#define HD    256          // per-direction hidden
#define G4    1024         // 4*HD gate width
#define H_DIM 512
#define T_TAGS 5
#define START_TAG 3
#define STOP_TAG  4
#define NEG_VAL  (-10000.0f)

typedef __attribute__((ext_vector_type(2))) float v2f;
typedef __attribute__((ext_vector_type(8))) float v8f;

// =====================================================================
// Kernel 1: embedding gather + input projection  xg = x @ w_ih^T + b
// fp32 WMMA 16x16x4. One wave (32 threads) per 16x16 output tile.
// A layout (32b 16x4): lane l -> M=l&15, VGPR v -> K = (l>>4)*2 + v
// B layout (32b 4x16): lane l -> N=l&15, VGPR v -> K = (l>>4)*2 + v
// C/D layout (32b 16x16): lane l -> N=l&15, VGPR v -> M = (l>>4)*8 + v
// =====================================================================
__global__ __launch_bounds__(32) void gather_proj_wmma(
    const int* __restrict__ sent, const float* __restrict__ embed,
    const float* __restrict__ w_ih_f, const float* __restrict__ b_f,
    const float* __restrict__ w_ih_b, const float* __restrict__ b_b,
    float* __restrict__ xg_f, float* __restrict__ xg_b) {
  const int lane = threadIdx.x;
  const int s0   = blockIdx.x * 16;      // sequence tile
  const int n0   = blockIdx.y * 16;      // gate-unit tile
  const int dir  = blockIdx.z;

  const float* w    = dir ? w_ih_b : w_ih_f;
  const float* bias = dir ? b_b    : b_f;
  float*       xg   = dir ? xg_b   : xg_f;

  const int m    = lane & 15;
  const int koff = (lane >> 4) * 2;

  const int tok = sent[s0 + m];
  const float* arow = embed + (size_t)tok * E_DIM;          // A row M=m
  const float* brow = w + (size_t)(n0 + m) * E_DIM;         // B[K][N]=w[N][K]

  const float bv = bias[n0 + m];                            // depends on N only
  v8f acc;
  #pragma unroll
  for (int v = 0; v < 8; ++v) acc[v] = bv;

  #pragma unroll 8
  for (int k0 = 0; k0 < E_DIM; k0 += 4) {
    v2f a = *(const v2f*)(arow + k0 + koff);
    v2f b = *(const v2f*)(brow + k0 + koff);
    // (neg_a, A, neg_b, B, c_mod, C, reuse_a, reuse_b)
    acc = __builtin_amdgcn_wmma_f32_16x16x4_f32(
        false, a, false, b, (short)0, acc, false, false);
  }

  const int col   = n0 + m;
  const int rbase = s0 + ((lane >> 4) * 8);
  #pragma unroll
  for (int v = 0; v < 8; ++v)
    xg[(size_t)(rbase + v) * G4 + col] = acc[v];
}

// =====================================================================
// Kernel 2: sequential BiLSTM recurrence.
// 8 blocks: block = (dir<<2)|gate. Each block keeps its 256x256 fp32
// w_hh gate slice TRANSPOSED in LDS (stride 257 -> bank-conflict-free),
// using CDNA5's 320KB/WGP LDS. Gate blocks 0..2 publish sigma(i),
// sigma(f), tanh(g); gate 3 owns c, combines, broadcasts h.
// Cross-WG sync: monotonic flags, agent-scope release/acquire.
// =====================================================================
#define LSTRIDE 257

__device__ __forceinline__ float sigm(float x) {
  return 1.0f / (1.0f + __expf(-x));
}

__global__ __launch_bounds__(256) void lstm_rec(
    const float* __restrict__ w_hh_f, const float* __restrict__ w_hh_b,
    const float* __restrict__ xg_f,  const float* __restrict__ xg_b,
    float* __restrict__ hf, float* __restrict__ hb,
    float* __restrict__ xch,   // [2 dirs][4 slots][HD]; slot3 = h broadcast
    int*   __restrict__ flags  // [2 dirs][4]; 0..2 gate flags, 3 = h flag
) {
  __shared__ float wT[HD * LSTRIDE];   // 263KB: CDNA5-only (320KB LDS/WGP)
  __shared__ float hbuf[HD];

  const int dir  = blockIdx.x >> 2;
  const int gate = blockIdx.x & 3;
  const int tid  = threadIdx.x;

  const float* whh = dir ? w_hh_b : w_hh_f;
  const float* xg  = dir ? xg_b   : xg_f;
  float*       hs  = dir ? hb     : hf;
  float* myxch = xch + (size_t)(dir * 4 + gate) * HD;
  float* xh    = xch + (size_t)(dir * 4 + 3) * HD;
  int*   fl    = flags + dir * 4;

  // Preload transposed weights: wT[k*LSTRIDE + j] = w_hh[gate*HD+j][k]
  // global reads coalesced over k=tid; LDS writes (tid*257+e)%64 conflict-free
  for (int e = 0; e < HD; ++e)
    wT[tid * LSTRIDE + e] = whh[(size_t)(gate * HD + e) * HD + tid];
  __syncthreads();

  float c = 0.0f;   // only meaningful in gate-3 block

  for (int t = 0; t < S_LEN; ++t) {
    const int tt = dir ? (S_LEN - 1 - t) : t;

    // -- wait for h^{t-1} (h flag reaches t; t=0 trivially passes, xh zeroed)
    if (tid == 0) {
      while (__hip_atomic_load(&fl[3], __ATOMIC_ACQUIRE,
                               __HIP_MEMORY_SCOPE_AGENT) < t)
        __builtin_amdgcn_s_sleep(1);
    }
    __syncthreads();
    __builtin_amdgcn_fence(__ATOMIC_ACQUIRE, "agent");  // invalidate L0

    hbuf[tid] = xh[tid];
    __syncthreads();

    // -- matvec for this block's gate from LDS-resident weights
    float g = xg[(size_t)tt * G4 + gate * HD + tid];
    #pragma unroll 8
    for (int k = 0; k < HD; ++k)
      g = fmaf(wT[k * LSTRIDE + tid], hbuf[k], g);

    if (gate < 3) {
      const float val = (gate == 2) ? tanhf(g) : sigm(g);
      myxch[tid] = val;
      __threadfence();
      __syncthreads();
      if (tid == 0)
        __hip_atomic_store(&fl[gate], t + 1, __ATOMIC_RELEASE,
                           __HIP_MEMORY_SCOPE_AGENT);
      __syncthreads();
    } else {
      const float o = sigm(g);
      if (tid < 3) {
        while (__hip_atomic_load(&fl[tid], __ATOMIC_ACQUIRE,
                                 __HIP_MEMORY_SCOPE_AGENT) < t + 1)
          __builtin_amdgcn_s_sleep(1);
      }
      __syncthreads();
      __builtin_amdgcn_fence(__ATOMIC_ACQUIRE, "agent");
      const float si = xch[(size_t)(dir * 4 + 0) * HD + tid];
      const float sf = xch[(size_t)(dir * 4 + 1) * HD + tid];
      const float tg = xch[(size_t)(dir * 4 + 2) * HD + tid];
      c = sf * c + si * tg;
      const float h = o * tanhf(c);
      hs[(size_t)tt * HD + tid] = h;
      xh[tid] = h;
      __threadfence();
      __syncthreads();
      if (tid == 0)
        __hip_atomic_store(&fl[3], t + 1, __ATOMIC_RELEASE,
                           __HIP_MEMORY_SCOPE_AGENT);
      __syncthreads();
    }
  }
}

// =====================================================================
// Kernel 3: feats[t] = [hf[t], hb[t]] @ W_tag^T + b_tag.  One wave per t.
// =====================================================================
__global__ __launch_bounds__(32) void tag_proj(
    const float* __restrict__ hf, const float* __restrict__ hb,
    const float* __restrict__ W_tag, const float* __restrict__ b_tag,
    float* __restrict__ feats) {
  const int t = blockIdx.x, lane = threadIdx.x;
  float acc[T_TAGS] = {0.f, 0.f, 0.f, 0.f, 0.f};
  for (int k = lane; k < HD; k += 32) {
    const float hv = hf[(size_t)t * HD + k];
    #pragma unroll
    for (int r = 0; r < T_TAGS; ++r)
      acc[r] = fmaf(hv, W_tag[r * H_DIM + k], acc[r]);
  }
  for (int k = lane; k < HD; k += 32) {
    const float hv = hb[(size_t)t * HD + k];
    #pragma unroll
    for (int r = 0; r < T_TAGS; ++r)
      acc[r] = fmaf(hv, W_tag[r * H_DIM + HD + k], acc[r]);
  }
  #pragma unroll
  for (int off = 16; off > 0; off >>= 1)
    #pragma unroll
    for (int r = 0; r < T_TAGS; ++r)
      acc[r] += __shfl_down(acc[r], off, 32);
  if (lane == 0)
    #pragma unroll
    for (int r = 0; r < T_TAGS; ++r)
      feats[(size_t)t * 8 + r] = acc[r] + b_tag[r];
}

// =====================================================================
// Kernel 4: Viterbi. One wave. Lanes 0..24 score (next,prev) pairs,
// lanes 0..4 reduce (first-max to match jnp.argmax), lane 0 backtracks.
// =====================================================================
__global__ __launch_bounds__(32) void viterbi_k(
    const float* __restrict__ feats, const float* __restrict__ trans,
    int* __restrict__ bptr, float* __restrict__ out) {
  __shared__ float fv[8];
  __shared__ float sbuf[32];
  __shared__ float tr[32];
  const int l = threadIdx.x;
  if (l < 25) tr[l] = trans[l];                 // row-major [next][prev]
  if (l < T_TAGS) fv[l] = (l == START_TAG) ? 0.0f : NEG_VAL;
  __syncthreads();

  for (int t = 0; t < S_LEN; ++t) {
    float s = -3.0e38f;
    if (l < 25) s = fv[l % T_TAGS] + tr[l];     // l = next*5 + prev
    sbuf[l] = s;
    __syncthreads();
    float nfv = 0.0f;
    if (l < T_TAGS) {
      float best = sbuf[l * T_TAGS]; int arg = 0;
      #pragma unroll
      for (int p = 1; p < T_TAGS; ++p) {
        const float v = sbuf[l * T_TAGS + p];
        if (v > best) { best = v; arg = p; }    // first-max tie-break
      }
      bptr[t * T_TAGS + l] = arg;
      nfv = best + feats[(size_t)t * 8 + l];
    }
    __syncthreads();
    if (l < T_TAGS) fv[l] = nfv;
    __syncthreads();
  }

  if (l == 0) {
    float best = fv[0] + tr[STOP_TAG * T_TAGS + 0]; int tag = 0;
    for (int p = 1; p < T_TAGS; ++p) {
      const float v = fv[p] + tr[STOP_TAG * T_TAGS + p];
      if (v > best) { best = v; tag = p; }
    }
    out[0] = best;                               // score
    for (int t = S_LEN - 1; t >= 0; --t) {       // path[t]=tag; tag=bp[t][tag]
      out[1 + t] = (float)tag;
      tag = bptr[t * T_TAGS + tag];
    }
  }
}

// zero the sync flags + exchange buffer every launch (graph-replay safe)
__global__ void init_k(float* __restrict__ xch, int* __restrict__ flags) {
  const int i = blockIdx.x * blockDim.x + threadIdx.x;
  if (i < 2 * 4 * HD) xch[i] = 0.0f;
  if (i < 16) flags[i] = 0;
}

// =====================================================================
extern "C" void kernel_launch(void* const* d_in, const int* in_sizes, int n_in,
                              void* d_out, int out_size, void* d_ws, size_t ws_size,
                              hipStream_t stream) {
  const int*   sent   = (const int*)  d_in[0];
  const float* embed  = (const float*)d_in[1];
  const float* w_ih_f = (const float*)d_in[2];
  const float* w_hh_f = (const float*)d_in[3];
  const float* b_f    = (const float*)d_in[4];
  const float* w_ih_b = (const float*)d_in[5];
  const float* w_hh_b = (const float*)d_in[6];
  const float* b_b    = (const float*)d_in[7];
  const float* W_tag  = (const float*)d_in[8];
  const float* b_tag  = (const float*)d_in[9];
  const float* trans  = (const float*)d_in[10];
  float* out = (float*)d_out;

  float* ws = (float*)d_ws;
  size_t o = 0;
  float* xg_f  = ws + o; o += (size_t)S_LEN * G4;
  float* xg_b  = ws + o; o += (size_t)S_LEN * G4;
  float* hf    = ws + o; o += (size_t)S_LEN * HD;
  float* hb    = ws + o; o += (size_t)S_LEN * HD;
  float* feats = ws + o; o += (size_t)S_LEN * 8;
  float* xch   = ws + o; o += 2 * 4 * HD;
  int*   flags = (int*)(ws + o); o += 16;
  int*   bptr  = (int*)(ws + o); o += (size_t)S_LEN * T_TAGS;

  init_k<<<2, 1024, 0, stream>>>(xch, flags);

  gather_proj_wmma<<<dim3(S_LEN / 16, G4 / 16, 2), 32, 0, stream>>>(
      sent, embed, w_ih_f, b_f, w_ih_b, b_b, xg_f, xg_b);

  lstm_rec<<<8, 256, 0, stream>>>(w_hh_f, w_hh_b, xg_f, xg_b,
                                  hf, hb, xch, flags);

  tag_proj<<<S_LEN, 32, 0, stream>>>(hf, hb, W_tag, b_tag, feats);

  viterbi_k<<<1, 32, 0, stream>>>(feats, trans, bptr, out);
}